// QIntSoftmax_74560632259333
// MI455X (gfx1250) — compile-verified
//
#include <hip/hip_runtime.h>
#include <hip/hip_bf16.h>

// ---------------------------------------------------------------------------
// QIntSoftmax for MI455X (gfx1250).
// Memory-bound (AI ~2 FLOP/B): one b128 global read + one b128 global store
// per 4 elements, rows staged through LDS with the CDNA5 async data mover
// (global_load_async_to_lds_b128 + ASYNCcnt), double-buffered so row i+1
// streams in while row i is reduced/normalized. wave32-aware reductions.
// ---------------------------------------------------------------------------

#define TPB 256            // 8 wave32 per block
#define VEC 4              // floats per thread (one b128)
#define SEQ 1024           // row length (S)
#define ROWS_PER_BLOCK 8   // rows pipelined per block
#define NWAVES (TPB / 32)

typedef int v4i __attribute__((vector_size(16)));   // matches builtin param type

#if defined(__HIP_DEVICE_COMPILE__) && __has_builtin(__builtin_amdgcn_global_load_async_to_lds_b128)
#define HAVE_ASYNC_LDS 1
#else
#define HAVE_ASYNC_LDS 0
#endif

__device__ __forceinline__ void async_row_load(const float* gsrc, float* lds_dst) {
#if HAVE_ASYNC_LDS
    // one 16B async transfer per lane: 256 lanes x 16B = 4KB row
    __builtin_amdgcn_global_load_async_to_lds_b128(
        (v4i*)(gsrc), (v4i*)(lds_dst), 0, /*cpol=*/0);
#else
    // fallback: synchronous copy through registers
    *(float4*)lds_dst = *(const float4*)gsrc;
#endif
}

__device__ __forceinline__ void wait_async_le1() {
#if HAVE_ASYNC_LDS
#if __has_builtin(__builtin_amdgcn_s_wait_asynccnt)
    __builtin_amdgcn_s_wait_asynccnt(1);
#else
    asm volatile("s_wait_asynccnt 0x1" ::: "memory");
#endif
#endif
}

__device__ __forceinline__ void wait_async_le0() {
#if HAVE_ASYNC_LDS
#if __has_builtin(__builtin_amdgcn_s_wait_asynccnt)
    __builtin_amdgcn_s_wait_asynccnt(0);
#else
    asm volatile("s_wait_asynccnt 0x0" ::: "memory");
#endif
#endif
}

// two-level reduction: wave32 xor-shuffle tree, then LDS across 8 waves
template <bool IS_MAX>
__device__ __forceinline__ float block_reduce(float v, float* red, int lane, int wave) {
#pragma unroll
    for (int off = 16; off > 0; off >>= 1) {
        float o = __shfl_xor(v, off, 32);
        v = IS_MAX ? fmaxf(v, o) : (v + o);
    }
    if (lane == 0) red[wave] = v;
    __syncthreads();
    float r = red[0];
#pragma unroll
    for (int w = 1; w < NWAVES; ++w)
        r = IS_MAX ? fmaxf(r, red[w]) : (r + red[w]);
    __syncthreads();
    return r;
}

__global__ __launch_bounds__(TPB) void qint_softmax_kernel(
    const float* __restrict__ x,
    const float* __restrict__ scales,
    float* __restrict__ out,
    int nrows, int H) {

    __shared__ alignas(16) float buf[2][SEQ];
    __shared__ float red[NWAVES];

    const int t    = threadIdx.x;
    const int lane = t & 31;
    const int wave = t >> 5;

    const long long rowBase = (long long)blockIdx.x * ROWS_PER_BLOCK;
    if (rowBase >= nrows) return;

    // per-block scalars (head index constant across the block's 8 rows since
    // ROWS_PER_BLOCK divides SEQ)
    const int h = (int)((rowBase / SEQ) % H);
    const float s      = scales[h];
    const float inv_s  = 1.0f / s;
    const float x0_int = floorf(-0.6931f / s);                        // floor(X0/s)
    const float inv_x0 = 1.0f / x0_int;
    const float nx0    = 30.0f * x0_int;                              // N_BITS * x0_int
    const float b_int  = floorf((float)(0.96963238 / 0.35815147) / s);
    const float c_int  = floorf((float)(1.0 / 0.35815147) / (s * s));

    // ---- pipeline prologue: async-load row 0 into buf[0] ----
    {
        const float* g0 = x + rowBase * SEQ + (size_t)t * VEC;
        async_row_load(g0, &buf[0][t * VEC]);
    }

    for (int i = 0; i < ROWS_PER_BLOCK; ++i) {
        const long long row = rowBase + i;
        const bool haveNext = (i + 1 < ROWS_PER_BLOCK) && (row + 1 < nrows);

        // issue async load of the next row into the other buffer
        if (haveNext) {
            const float* gn = x + (row + 1) * SEQ + (size_t)t * VEC;
            async_row_load(gn, &buf[(i + 1) & 1][t * VEC]);
            wait_async_le1();   // retire row i's transfer (in-order completion)
        } else {
            wait_async_le0();
        }
        __syncthreads();        // all waves' LDS slices of row i visible

        if (row < nrows) {
            // pull this thread's 4 elements out of LDS (ds_load_b128)
            const float4 v = ((const float4*)buf[i & 1])[t];

            const float xi0 = v.x * inv_s;
            const float xi1 = v.y * inv_s;
            const float xi2 = v.z * inv_s;
            const float xi3 = v.w * inv_s;

            float m = fmaxf(fmaxf(xi0, xi1), fmaxf(xi2, xi3));
            const float M = block_reduce<true>(m, red, lane, wave);

            float e[VEC];
            const float xin[VEC] = {xi0, xi1, xi2, xi3};
            float lsum = 0.0f;
#pragma unroll
            for (int k = 0; k < VEC; ++k) {
                float xc = fmaxf(xin[k] - M, nx0);          // clamp at N_BITS*x0
                float q  = floorf(xc * inv_x0);             // quotient (0..30)
                float r  = xc - x0_int * q;                 // remainder
                float z  = r * (r + b_int) + c_int;         // quadratic exp approx
                float ex = __builtin_ldexpf(z, 30 - (int)q);// z * 2^(N_BITS - q)
                ex = fmaxf(floorf(ex), 0.0f);
                e[k] = ex;
                lsum += ex;
            }

            const float SUM = block_reduce<false>(lsum, red, lane, wave);
            const float inv_sum = 1.0f / SUM;

            float4 o;
            o.x = e[0] * inv_sum;
            o.y = e[1] * inv_sum;
            o.z = e[2] * inv_sum;
            o.w = e[3] * inv_sum;
            ((float4*)(out + row * SEQ))[t] = o;            // global_store_b128
        }
        // trailing barrier of block_reduce<false> guarantees no thread still
        // reads buf[i&1] when iteration i+1 re-issues into it
    }
}

extern "C" void kernel_launch(void* const* d_in, const int* in_sizes, int n_in,
                              void* d_out, int out_size, void* d_ws, size_t ws_size,
                              hipStream_t stream) {
    const float* x      = (const float*)d_in[0];
    const float* scales = (const float*)d_in[1];
    float*       out    = (float*)d_out;

    const long long total = (long long)in_sizes[0];
    const int nrows = (int)(total / SEQ);       // 8*12*1024 = 98304
    const int H     = (n_in > 1) ? in_sizes[1] : 12;

    const int blocks = (nrows + ROWS_PER_BLOCK - 1) / ROWS_PER_BLOCK;  // 12288
    qint_softmax_kernel<<<blocks, TPB, 0, stream>>>(x, scales, out, nrows, H);
}